// FraudGAT_55061480734903
// MI455X (gfx1250) — compile-verified
//
#include <hip/hip_runtime.h>
#include <math.h>

// ---------------------------------------------------------------------------
// FraudGAT (2-layer GAT) for gfx1250 / MI455X.
//   L1: h = x @ W1 (WMMA f32 16x16x4), per-head attention softmax over edges,
//       message scatter with f32 atomics (L2-resident), bias + ELU.
//   L2: 256->2 linear (VALU), 1-head attention, scatter, bias.
// ---------------------------------------------------------------------------

#define N_NODES   50000
#define IN_DIM    128
#define HID       64
#define HEADS     4
#define HC        (HEADS * HID)   // 256
#define OUT_DIM   2
#define NEG_SLOPE 0.2f

typedef float v2f __attribute__((ext_vector_type(2)));
typedef float v8f __attribute__((ext_vector_type(8)));

// ---------------- utility ----------------

__global__ void fill_f32(float* __restrict__ p, float v, int n) {
    int t = blockIdx.x * blockDim.x + threadIdx.x;
    if (t < n) p[t] = v;
}

__device__ __forceinline__ void atomicMaxFloat(float* addr, float val) {
    // monotonic int/uint reinterpretation trick
    if (val >= 0.0f)
        atomicMax((int*)addr, __float_as_int(val));
    else
        atomicMin((unsigned int*)addr, __float_as_uint(val));
}

__device__ __forceinline__ float leaky(float v) {
    return v > 0.0f ? v : NEG_SLOPE * v;
}

// ---------------- layer 1: GEMM via fp32 WMMA ----------------
// H[N,256] = X[N,128] @ W[128,256]; one wave per 16x16 output tile.
// M tiles = 3125, N tiles = 16 -> 50000 waves -> 6250 blocks x 8 waves.

__global__ __launch_bounds__(256) void gemm1_wmma(const float* __restrict__ X,
                                                  const float* __restrict__ W,
                                                  float* __restrict__ H) {
    const int wave  = (int)((blockIdx.x * blockDim.x + threadIdx.x) >> 5);
    const int lane  = threadIdx.x & 31;
    const int tm    = wave >> 4;        // 0..3124
    const int tn    = wave & 15;        // 0..15
    if (tm >= (N_NODES / 16)) return;   // never taken (exact tiling), wave-uniform

    const int l15   = lane & 15;
    const int khalf = lane >> 4;        // 0: K={k0,k0+1}, 1: K={k0+2,k0+3}
    const int row   = tm * 16 + l15;    // A row for this lane
    const int col   = tn * 16 + l15;    // B/C column for this lane

    const float* xrow = X + (size_t)row * IN_DIM + khalf * 2;

    v8f c = {};
    #pragma unroll 4
    for (int k0 = 0; k0 < IN_DIM; k0 += 4) {
        v2f a;
        a.x = xrow[k0];
        a.y = xrow[k0 + 1];
        const int kk = k0 + khalf * 2;
        v2f b;
        b.x = W[(size_t)kk * HC + col];
        b.y = W[(size_t)(kk + 1) * HC + col];
        // D = A(16x4) * B(4x16) + C  — v_wmma_f32_16x16x4_f32
        c = __builtin_amdgcn_wmma_f32_16x16x4_f32(
                /*neg_a=*/false, a, /*neg_b=*/false, b,
                /*c_mod=*/(short)0, c, /*reuse_a=*/false, /*reuse_b=*/false);
    }

    // C layout: VGPR i -> M = i (lanes 0-15) / M = 8+i (lanes 16-31), N = lane&15
    const int rbase = tm * 16 + khalf * 8;
    #pragma unroll
    for (int i = 0; i < 8; ++i)
        H[(size_t)(rbase + i) * HC + col] = c[i];
}

// per-(node,head) attention dot products: as/ad [N*HEADS]
__global__ void attn_dots1(const float* __restrict__ H,
                           const float* __restrict__ att_src,
                           const float* __restrict__ att_dst,
                           float* __restrict__ as1, float* __restrict__ ad1) {
    int t = blockIdx.x * blockDim.x + threadIdx.x;
    if (t >= N_NODES * HEADS) return;
    const int node = t >> 2, head = t & 3;
    const float* hp = H + (size_t)node * HC + head * HID;
    const float* sp = att_src + head * HID;
    const float* dp = att_dst + head * HID;
    float accs = 0.f, accd = 0.f;
    #pragma unroll 8
    for (int i = 0; i < HID; ++i) {
        const float v = hp[i];
        accs += v * sp[i];
        accd += v * dp[i];
    }
    as1[t] = accs;
    ad1[t] = accd;
}

__device__ __forceinline__ void edge_nodes(const int* ei, int e, int E,
                                           int& s, int& d) {
    if (e < E) { s = ei[e]; d = ei[E + e]; }
    else       { s = d = e - E; }          // self loops appended
}

// pass 1: segment max of leaky-relu logits (per dst, per head)
__global__ void edge_max1(const int* __restrict__ ei,
                          const float* __restrict__ as1,
                          const float* __restrict__ ad1,
                          float* __restrict__ m1, int E, int Etot) {
    int t = blockIdx.x * blockDim.x + threadIdx.x;
    if (t >= Etot * HEADS) return;
    const int e = t >> 2, head = t & 3;
    int s, d; edge_nodes(ei, e, E, s, d);
    const float v = leaky(as1[s * HEADS + head] + ad1[d * HEADS + head]);
    atomicMaxFloat(&m1[d * HEADS + head], v);
}

// pass 2: segment sum of exp(logit - max)
__global__ void edge_sum1(const int* __restrict__ ei,
                          const float* __restrict__ as1,
                          const float* __restrict__ ad1,
                          const float* __restrict__ m1,
                          float* __restrict__ z1, int E, int Etot) {
    int t = blockIdx.x * blockDim.x + threadIdx.x;
    if (t >= Etot * HEADS) return;
    const int e = t >> 2, head = t & 3;
    int s, d; edge_nodes(ei, e, E, s, d);
    const int di = d * HEADS + head;
    const float v = leaky(as1[s * HEADS + head] + ad1[di]);
    atomicAdd(&z1[di], __expf(v - m1[di]));
}

// pass 3: scatter alpha * h[src] into acc[dst]; 64 threads/edge, float4 each
__global__ void scatter1(const int* __restrict__ ei,
                         const float* __restrict__ as1,
                         const float* __restrict__ ad1,
                         const float* __restrict__ m1,
                         const float* __restrict__ z1,
                         const float* __restrict__ H,
                         float* __restrict__ acc, int E, int Etot) {
    int t = blockIdx.x * blockDim.x + threadIdx.x;
    if (t >= Etot * (HC / 4)) return;
    const int e = t >> 6;
    const int j = t & 63;        // float4 group -> channels [4j, 4j+3]
    const int head = j >> 4;     // 64 channels per head
    int s, d; edge_nodes(ei, e, E, s, d);
    const int di = d * HEADS + head;
    const float v = leaky(as1[s * HEADS + head] + ad1[di]);
    const float alpha = __expf(v - m1[di]) / z1[di];
    const float4 hv = *(const float4*)(H + (size_t)s * HC + j * 4);
    float* ap = acc + (size_t)d * HC + j * 4;
    atomicAdd(ap + 0, hv.x * alpha);
    atomicAdd(ap + 1, hv.y * alpha);
    atomicAdd(ap + 2, hv.z * alpha);
    atomicAdd(ap + 3, hv.w * alpha);
}

// bias + ELU in place
__global__ void bias_elu1(float* __restrict__ acc, const float* __restrict__ b) {
    int t = blockIdx.x * blockDim.x + threadIdx.x;
    if (t >= N_NODES * HC) return;
    const float v = acc[t] + b[t & (HC - 1)];
    acc[t] = v > 0.0f ? v : (__expf(v) - 1.0f);
}

// ---------------- layer 2 ----------------

__global__ void lin2(const float* __restrict__ act, const float* __restrict__ W2,
                     const float* __restrict__ asw, const float* __restrict__ adw,
                     float* __restrict__ h2, float* __restrict__ as2,
                     float* __restrict__ ad2) {
    int n = blockIdx.x * blockDim.x + threadIdx.x;
    if (n >= N_NODES) return;
    const float* a = act + (size_t)n * HC;
    float o0 = 0.f, o1 = 0.f;
    #pragma unroll 8
    for (int k = 0; k < HC; ++k) {
        const float v = a[k];
        o0 += v * W2[k * OUT_DIM + 0];
        o1 += v * W2[k * OUT_DIM + 1];
    }
    h2[n * OUT_DIM + 0] = o0;
    h2[n * OUT_DIM + 1] = o1;
    as2[n] = o0 * asw[0] + o1 * asw[1];
    ad2[n] = o0 * adw[0] + o1 * adw[1];
}

__global__ void edge_max2(const int* __restrict__ ei,
                          const float* __restrict__ as2,
                          const float* __restrict__ ad2,
                          float* __restrict__ m2, int E, int Etot) {
    int e = blockIdx.x * blockDim.x + threadIdx.x;
    if (e >= Etot) return;
    int s, d; edge_nodes(ei, e, E, s, d);
    atomicMaxFloat(&m2[d], leaky(as2[s] + ad2[d]));
}

__global__ void edge_sum2(const int* __restrict__ ei,
                          const float* __restrict__ as2,
                          const float* __restrict__ ad2,
                          const float* __restrict__ m2,
                          float* __restrict__ z2, int E, int Etot) {
    int e = blockIdx.x * blockDim.x + threadIdx.x;
    if (e >= Etot) return;
    int s, d; edge_nodes(ei, e, E, s, d);
    const float v = leaky(as2[s] + ad2[d]);
    atomicAdd(&z2[d], __expf(v - m2[d]));
}

__global__ void scatter2(const int* __restrict__ ei,
                         const float* __restrict__ as2,
                         const float* __restrict__ ad2,
                         const float* __restrict__ m2,
                         const float* __restrict__ z2,
                         const float* __restrict__ h2,
                         float* __restrict__ out, int E, int Etot) {
    int e = blockIdx.x * blockDim.x + threadIdx.x;
    if (e >= Etot) return;
    int s, d; edge_nodes(ei, e, E, s, d);
    const float v = leaky(as2[s] + ad2[d]);
    const float alpha = __expf(v - m2[d]) / z2[d];
    atomicAdd(&out[d * OUT_DIM + 0], h2[s * OUT_DIM + 0] * alpha);
    atomicAdd(&out[d * OUT_DIM + 1], h2[s * OUT_DIM + 1] * alpha);
}

__global__ void bias2(float* __restrict__ out, const float* __restrict__ b) {
    int t = blockIdx.x * blockDim.x + threadIdx.x;
    if (t >= N_NODES * OUT_DIM) return;
    out[t] += b[t & (OUT_DIM - 1)];
}

// ---------------- launch ----------------

static inline int cdiv(long long a, int b) { return (int)((a + b - 1) / b); }

extern "C" void kernel_launch(void* const* d_in, const int* in_sizes, int n_in,
                              void* d_out, int out_size, void* d_ws, size_t ws_size,
                              hipStream_t stream) {
    const float* x        = (const float*)d_in[0];
    const int*   ei       = (const int*)d_in[1];   // int32 (JAX coerces int64->int32)
    const float* W1       = (const float*)d_in[2];
    const float* att_src1 = (const float*)d_in[3];
    const float* att_dst1 = (const float*)d_in[4];
    const float* b1       = (const float*)d_in[5];
    const float* W2       = (const float*)d_in[6];
    const float* att_src2 = (const float*)d_in[7];
    const float* att_dst2 = (const float*)d_in[8];
    const float* b2       = (const float*)d_in[9];
    float* out = (float*)d_out;

    const int E    = in_sizes[1] / 2;      // 800000
    const int Etot = E + N_NODES;          // + self loops

    // workspace carve-up (256B aligned)
    char* w = (char*)d_ws;
    size_t off = 0;
    auto alloc = [&](size_t bytes) {
        float* p = (float*)(w + off);
        off += (bytes + 255) & ~(size_t)255;
        return p;
    };
    float* h1   = alloc((size_t)N_NODES * HC * 4);
    float* acc1 = alloc((size_t)N_NODES * HC * 4);   // also holds ELU(act)
    float* as1  = alloc((size_t)N_NODES * HEADS * 4);
    float* ad1  = alloc((size_t)N_NODES * HEADS * 4);
    float* m1   = alloc((size_t)N_NODES * HEADS * 4);
    float* z1   = alloc((size_t)N_NODES * HEADS * 4);
    float* h2   = alloc((size_t)N_NODES * OUT_DIM * 4);
    float* as2  = alloc((size_t)N_NODES * 4);
    float* ad2  = alloc((size_t)N_NODES * 4);
    float* m2   = alloc((size_t)N_NODES * 4);
    float* z2   = alloc((size_t)N_NODES * 4);
    (void)ws_size;

    const int B = 256;

    // init accumulators / sentinels (deterministic each call)
    fill_f32<<<cdiv(N_NODES * HC, B), B, 0, stream>>>(acc1, 0.0f, N_NODES * HC);
    fill_f32<<<cdiv(N_NODES * HEADS, B), B, 0, stream>>>(m1, -INFINITY, N_NODES * HEADS);
    fill_f32<<<cdiv(N_NODES * HEADS, B), B, 0, stream>>>(z1, 0.0f, N_NODES * HEADS);
    fill_f32<<<cdiv(N_NODES, B), B, 0, stream>>>(m2, -INFINITY, N_NODES);
    fill_f32<<<cdiv(N_NODES, B), B, 0, stream>>>(z2, 0.0f, N_NODES);
    fill_f32<<<cdiv(N_NODES * OUT_DIM, B), B, 0, stream>>>(out, 0.0f, N_NODES * OUT_DIM);

    // ----- layer 1 -----
    // 3125 * 16 = 50000 tiles; 8 waves per 256-thread block -> 6250 blocks
    gemm1_wmma<<<(N_NODES / 16) * (HC / 16) / 8, B, 0, stream>>>(x, W1, h1);
    attn_dots1<<<cdiv(N_NODES * HEADS, B), B, 0, stream>>>(h1, att_src1, att_dst1, as1, ad1);
    edge_max1<<<cdiv((long long)Etot * HEADS, B), B, 0, stream>>>(ei, as1, ad1, m1, E, Etot);
    edge_sum1<<<cdiv((long long)Etot * HEADS, B), B, 0, stream>>>(ei, as1, ad1, m1, z1, E, Etot);
    scatter1<<<cdiv((long long)Etot * (HC / 4), B), B, 0, stream>>>(ei, as1, ad1, m1, z1, h1, acc1, E, Etot);
    bias_elu1<<<cdiv(N_NODES * HC, B), B, 0, stream>>>(acc1, b1);

    // ----- layer 2 -----
    lin2<<<cdiv(N_NODES, B), B, 0, stream>>>(acc1, W2, att_src2, att_dst2, h2, as2, ad2);
    edge_max2<<<cdiv(Etot, B), B, 0, stream>>>(ei, as2, ad2, m2, E, Etot);
    edge_sum2<<<cdiv(Etot, B), B, 0, stream>>>(ei, as2, ad2, m2, z2, E, Etot);
    scatter2<<<cdiv(Etot, B), B, 0, stream>>>(ei, as2, ad2, m2, z2, h2, out, E, Etot);
    bias2<<<cdiv(N_NODES * OUT_DIM, B), B, 0, stream>>>(out, b2);
}